// CascadeSSDLoss_27135603376432
// MI455X (gfx1250) — compile-verified
//
#include <hip/hip_runtime.h>
#include <hip/hip_bf16.h>

#define NP 524288          // NUM_PRIORS
#define NT 100             // NUM_TARGETS
#define NC 21              // NUM_CLASSES + 1 (background)
#define NBLK (NP / 256)    // 2048 blocks of 256 threads
#define TIE_CAP 1024

typedef __attribute__((ext_vector_type(2))) float v2f;
typedef __attribute__((ext_vector_type(8))) float v8f;
typedef __attribute__((__vector_size__(4 * sizeof(int)))) int v4i;

// ---------------------------------------------------------------------------
// CDNA5 async memory->LDS copy (ASYNCcnt path).
// Builtin signatures (from hipcc diagnostics on this toolchain):
//   __builtin_amdgcn_global_load_async_to_lds_b128(v4i* as1, v4i* as3, Ii, Ii)
//   __builtin_amdgcn_global_load_async_to_lds_b32 (int* as1, int* as3, Ii, Ii)
// Low 32 bits of a generic pointer to LDS are the LDS byte offset (aperture
// mapping).
// ---------------------------------------------------------------------------
#if __has_builtin(__builtin_amdgcn_global_load_async_to_lds_b128)
typedef __attribute__((address_space(1))) v4i g_v4i_t;
typedef __attribute__((address_space(3))) v4i l_v4i_t;
typedef __attribute__((address_space(1))) int g_i32_t;
typedef __attribute__((address_space(3))) int l_i32_t;
#define ASYNC_BUILTIN 1
#endif

__device__ __forceinline__ void async_copy_b128(const void* g, void* l) {
#ifdef ASYNC_BUILTIN
  __builtin_amdgcn_global_load_async_to_lds_b128(
      (g_v4i_t*)(uintptr_t)g,
      (l_v4i_t*)(unsigned)(uintptr_t)l, 0, 0);
#else
  unsigned lo = (unsigned)(uintptr_t)l;
  asm volatile("global_load_async_to_lds_b128 %0, %1, off"
               :: "v"(lo), "v"(g) : "memory");
#endif
}

__device__ __forceinline__ void async_copy_b32(const void* g, void* l) {
#ifdef ASYNC_BUILTIN
  __builtin_amdgcn_global_load_async_to_lds_b32(
      (g_i32_t*)(uintptr_t)g,
      (l_i32_t*)(unsigned)(uintptr_t)l, 0, 0);
#else
  unsigned lo = (unsigned)(uintptr_t)l;
  asm volatile("global_load_async_to_lds_b32 %0, %1, off"
               :: "v"(lo), "v"(g) : "memory");
#endif
}

__device__ __forceinline__ void async_wait0() {
#if __has_builtin(__builtin_amdgcn_s_wait_asynccnt)
  __builtin_amdgcn_s_wait_asynccnt(0);
#else
  asm volatile("s_wait_asynccnt 0x0" ::: "memory");
#endif
}

// ---------------------------------------------------------------------------
// Wave32 sum via V_WMMA_F32_16X16X4_F32:
// A(16x4): lane m (<16) supplies A[m][0..1]; lane m+16 supplies A[m][2..3].
// With a1 = 0 and B = ones(4x16): D[m][n] = v(lane m) + v(lane m+16).
// Sum of D VGPRs 0..7: lane n<16 -> sum_{m<8} rowsum(m); lane n+16 ->
// sum_{m>=8} rowsum(m).  lane0 + lane16 = full 32-lane sum (f32, fixed order).
// ---------------------------------------------------------------------------
__device__ __forceinline__ float wave_sum(float v) {
#if __has_builtin(__builtin_amdgcn_wmma_f32_16x16x4_f32)
  v2f a; a[0] = v;    a[1] = 0.0f;
  v2f b; b[0] = 1.0f; b[1] = 1.0f;
  v8f c = {};
  v8f d = __builtin_amdgcn_wmma_f32_16x16x4_f32(
      /*neg_a=*/false, a, /*neg_b=*/false, b,
      /*c_mod=*/(short)0, c, /*reuse_a=*/false, /*reuse_b=*/false);
  float s = d[0] + d[1] + d[2] + d[3] + d[4] + d[5] + d[6] + d[7];
  return __shfl(s, 0, 32) + __shfl(s, 16, 32);
#else
  for (int off = 16; off > 0; off >>= 1) v += __shfl_xor(v, off, 32);
  return v;
#endif
}

// Valid result only on threadIdx.x == 0.
__device__ __forceinline__ float block_sum256(float v, float* s8) {
  int tid = threadIdx.x;
  float ws = wave_sum(v);
  if ((tid & 31) == 0) s8[tid >> 5] = ws;
  __syncthreads();
  float tot = 0.0f;
  if (tid == 0)
    for (int q = 0; q < 8; ++q) tot += s8[q];
  return tot;
}

// ---------------------------------------------------------------------------
// K0: zero the bits of workspace that are accumulated into.
// ---------------------------------------------------------------------------
__global__ void k_init(unsigned long long* key64, unsigned* hist,
                       int* num_pos, unsigned* prefix, unsigned* krem,
                       int* tie_cnt) {
  int tid = threadIdx.x;
  if (tid < NT) key64[tid] = 0ull;
  hist[tid] = 0u;
  if (tid == 0) { *num_pos = 0; *prefix = 0u; *krem = 0u; *tie_cnt = 0; }
}

// ---------------------------------------------------------------------------
// K1: IoU row per prior: best target (first-max argmax) per prior, and
// per-target best prior via packed 64-bit atomicMax (ties -> lowest idx).
// gt_boxes staged to LDS with async b128 copies.
// ---------------------------------------------------------------------------
__global__ void k_iou(const float* __restrict__ priors,
                      const float* __restrict__ gt_boxes,
                      float* __restrict__ best_iou,
                      int* __restrict__ best_idx,
                      unsigned long long* __restrict__ key64) {
  __shared__ __align__(16) float s_gt[NT * 4];
  __shared__ unsigned long long s_key[NT];
  int tid = threadIdx.x;
  if (tid < NT) async_copy_b128(gt_boxes + (size_t)tid * 4, &s_gt[tid * 4]);
  for (int j = tid; j < NT; j += 256) s_key[j] = 0ull;
  async_wait0();
  __syncthreads();

  int i = blockIdx.x * 256 + tid;
  float cx = priors[(size_t)i * 4 + 0], cy = priors[(size_t)i * 4 + 1];
  float w  = priors[(size_t)i * 4 + 2], h  = priors[(size_t)i * 4 + 3];
  float px1 = cx - 0.5f * w, py1 = cy - 0.5f * h;
  float px2 = cx + 0.5f * w, py2 = cy + 0.5f * h;
  float area_p = fmaxf(px2 - px1, 0.0f) * fmaxf(py2 - py1, 0.0f);

  float best = -1.0f;
  int bidx = 0;
  unsigned inv_i = ~(unsigned)i;
  for (int t = 0; t < NT; ++t) {
    float gx1 = s_gt[t * 4 + 0], gy1 = s_gt[t * 4 + 1];
    float gx2 = s_gt[t * 4 + 2], gy2 = s_gt[t * 4 + 3];
    float ix1 = fmaxf(gx1, px1), iy1 = fmaxf(gy1, py1);
    float ix2 = fminf(gx2, px2), iy2 = fminf(gy2, py2);
    float ov = fmaxf(ix2 - ix1, 0.0f) * fmaxf(iy2 - iy1, 0.0f);
    float ag = fmaxf(gx2 - gx1, 0.0f) * fmaxf(gy2 - gy1, 0.0f);
    float iou = ov / (ag + area_p - ov + 1e-5f);
    if (iou > best) { best = iou; bidx = t; }  // first occurrence of max
    unsigned long long key =
        ((unsigned long long)__float_as_uint(iou) << 32) | (unsigned long long)inv_i;
    if (key > s_key[t]) atomicMax(&s_key[t], key);  // racy pre-check is fine
  }
  best_iou[i] = best;
  best_idx[i] = bidx;
  __syncthreads();
  for (int t = tid; t < NT; t += 256)
    if (s_key[t]) atomicMax(&key64[t], s_key[t]);
}

// ---------------------------------------------------------------------------
// K2: sequential override, j ascending (matches the JAX .at[].set order).
// ---------------------------------------------------------------------------
__global__ void k_override(const unsigned long long* __restrict__ key64,
                           float* __restrict__ best_iou,
                           int* __restrict__ best_idx) {
  if (blockIdx.x == 0 && threadIdx.x == 0) {
    for (int j = 0; j < NT; ++j) {
      unsigned p = ~(unsigned)(key64[j] & 0xFFFFFFFFull);
      best_idx[p] = j;
      best_iou[p] = 2.0f;
    }
  }
}

// ---------------------------------------------------------------------------
// K3: per-prior labels, log-softmax, mining key, CE, smooth-L1 for positives.
// gt tables staged via async copies; (reg + positive CE) reduced per block
// with the WMMA reduction; num_pos via integer atomics (deterministic).
// ---------------------------------------------------------------------------
__global__ void k_loss(const float* __restrict__ preds,
                       const float* __restrict__ gt_boxes,
                       const int* __restrict__ gt_labels,
                       const float* __restrict__ priors,
                       const float* __restrict__ best_iou,
                       const int* __restrict__ best_idx,
                       unsigned* __restrict__ mlkey,
                       float* __restrict__ ce_arr,
                       unsigned char* __restrict__ posflag,
                       float* __restrict__ partial_base,
                       int* __restrict__ num_pos_g) {
  __shared__ __align__(16) float s_gt[NT * 4];
  __shared__ __align__(16) int   s_lab[NT];
  __shared__ float s8[8];
  __shared__ int   s_cnt;
  int tid = threadIdx.x;
  if (tid < NT) {
    async_copy_b128(gt_boxes + (size_t)tid * 4, &s_gt[tid * 4]);
    async_copy_b32(gt_labels + tid, &s_lab[tid]);
  }
  if (tid == 0) s_cnt = 0;
  async_wait0();
  __syncthreads();

  int i = blockIdx.x * 256 + tid;
  const float* row = preds + (size_t)i * 25;
  if (i + 256 < NP) __builtin_prefetch(preds + (size_t)(i + 256) * 25, 0, 3);

  int t = best_idx[i];
  float biou = best_iou[i];
  int lab = (biou < 0.5f) ? 0 : s_lab[t];

  float c[NC];
  float m = -3.402823466e+38f;
  for (int j = 0; j < NC; ++j) { c[j] = row[4 + j]; m = fmaxf(m, c[j]); }
  float se = 0.0f;
  for (int j = 0; j < NC; ++j) se += __expf(c[j] - m);
  float lse = m + __logf(se);

  float mining = lse - c[0];       // -log_softmax[0] >= 0
  float ce = lse - c[lab];
  ce_arr[i] = ce;
  posflag[i] = (lab > 0) ? 1 : 0;
  mlkey[i] = (lab > 0) ? 0u : __float_as_uint(mining);

  float base = 0.0f;
  if (lab > 0) {
    float gx1 = s_gt[t * 4 + 0], gy1 = s_gt[t * 4 + 1];
    float gx2 = s_gt[t * 4 + 2], gy2 = s_gt[t * 4 + 3];
    float mcx = 0.5f * (gx1 + gx2), mcy = 0.5f * (gy1 + gy2);
    float mw = gx2 - gx1, mh = gy2 - gy1;
    float pcx = priors[(size_t)i * 4 + 0], pcy = priors[(size_t)i * 4 + 1];
    float pw  = priors[(size_t)i * 4 + 2], ph  = priors[(size_t)i * 4 + 3];
    float gl[4];
    gl[0] = (mcx - pcx) / pw / 0.1f;
    gl[1] = (mcy - pcy) / ph / 0.1f;
    gl[2] = __logf(mw / pw) / 0.2f;
    gl[3] = __logf(mh / ph) / 0.2f;
    float reg = 0.0f;
    for (int k = 0; k < 4; ++k) {
      float d = fabsf(row[k] - gl[k]);
      reg += (d < 1.0f) ? 0.5f * d * d : (d - 0.5f);
    }
    base = reg + ce;               // reg_loss and positive CE share one sum
    atomicAdd(&s_cnt, 1);
  }

  float tot = block_sum256(base, s8);   // wmma reduction; valid on tid 0
  if (tid == 0) {
    partial_base[blockIdx.x] = tot;
    atomicAdd(num_pos_g, s_cnt);
  }
}

// ---------------------------------------------------------------------------
// K4a: one radix byte histogram pass (MSB first) restricted to current prefix.
// ---------------------------------------------------------------------------
__global__ void k_hist(const unsigned* __restrict__ mlkey,
                       unsigned* __restrict__ hist,
                       const unsigned* __restrict__ prefix_p, int pass) {
  __shared__ unsigned h[256];
  int tid = threadIdx.x;
  h[tid] = 0u;
  __syncthreads();
  int i = blockIdx.x * 256 + tid;
  unsigned key = mlkey[i];
  int shift = 24 - 8 * pass;
  bool match = (pass == 0) ||
               ((key >> (shift + 8)) == (*prefix_p >> (shift + 8)));
  if (match) atomicAdd(&h[(key >> shift) & 0xFFu], 1u);
  __syncthreads();
  if (h[tid]) atomicAdd(&hist[tid], h[tid]);
}

// ---------------------------------------------------------------------------
// K4b: pick the byte bucket containing the K-th largest, update prefix/K,
// zero the histogram for the next pass.
// ---------------------------------------------------------------------------
__global__ void k_scan(unsigned* __restrict__ hist,
                       unsigned* __restrict__ prefix_p,
                       unsigned* __restrict__ krem_p,
                       const int* __restrict__ num_pos_g, int pass) {
  __shared__ unsigned h[256];
  int tid = threadIdx.x;
  h[tid] = hist[tid];
  hist[tid] = 0u;                 // ready for next pass
  __syncthreads();
  if (tid == 0) {
    unsigned K = (pass == 0) ? (unsigned)(3 * *num_pos_g) : *krem_p;
    long long cum = 0;
    int b = 0;
    for (int q = 255; q >= 0; --q) {
      cum += (long long)h[q];
      if (cum >= (long long)K) { b = q; break; }
    }
    unsigned newK = K - (unsigned)(cum - (long long)h[b]);
    *prefix_p = *prefix_p | ((unsigned)b << (24 - 8 * pass));
    *krem_p = newK;
  }
}

// ---------------------------------------------------------------------------
// K5: sum CE over strictly-above-threshold negatives (wmma block reduction,
// per-block partials), gather tie indices (key == T) for stable tie-break.
// ---------------------------------------------------------------------------
__global__ void k_select(const unsigned* __restrict__ mlkey,
                         const float* __restrict__ ce_arr,
                         const unsigned char* __restrict__ posflag,
                         const unsigned* __restrict__ prefix_p,
                         float* __restrict__ partial_neg,
                         int* __restrict__ tie_cnt,
                         int* __restrict__ tie_idx) {
  __shared__ float s8[8];
  int tid = threadIdx.x;
  int i = blockIdx.x * 256 + tid;
  unsigned T = *prefix_p;
  unsigned key = mlkey[i];
  bool neg = (posflag[i] == 0);
  float v = (neg && key > T) ? ce_arr[i] : 0.0f;
  if (neg && key == T) {
    int q = atomicAdd(tie_cnt, 1);
    if (q < TIE_CAP) tie_idx[q] = i;
  }
  float tot = block_sum256(v, s8);
  if (tid == 0) partial_neg[blockIdx.x] = tot;
}

// ---------------------------------------------------------------------------
// K6: fixed-order final reduction + stable tie resolution + normalize.
// ---------------------------------------------------------------------------
__global__ void k_final(const float* __restrict__ partial_base,
                        const float* __restrict__ partial_neg,
                        const float* __restrict__ ce_arr,
                        const int* __restrict__ num_pos_g,
                        const unsigned* __restrict__ krem_p,
                        const int* __restrict__ tie_cnt,
                        int* __restrict__ tie_idx,
                        float* __restrict__ out) {
  if (blockIdx.x != 0 || threadIdx.x != 0) return;
  float base = 0.0f, negs = 0.0f;
  for (int q = 0; q < NBLK; ++q) base += partial_base[q];
  for (int q = 0; q < NBLK; ++q) negs += partial_neg[q];

  int nt = *tie_cnt; if (nt > TIE_CAP) nt = TIE_CAP;
  // sort tie indices ascending (stable argsort takes lowest indices first)
  for (int a = 0; a < nt; ++a)
    for (int b = a + 1; b < nt; ++b)
      if (tie_idx[b] < tie_idx[a]) { int tmp = tie_idx[a]; tie_idx[a] = tie_idx[b]; tie_idx[b] = tmp; }
  int r = (int)*krem_p; if (r > nt) r = nt;
  float ts = 0.0f;
  for (int q = 0; q < r; ++q) ts += ce_arr[tie_idx[q]];

  int np = *num_pos_g;
  float denom = (float)((np > 1) ? np : 1);
  out[0] = (base + negs + ts) / denom;
}

// ---------------------------------------------------------------------------
extern "C" void kernel_launch(void* const* d_in, const int* in_sizes, int n_in,
                              void* d_out, int out_size, void* d_ws, size_t ws_size,
                              hipStream_t stream) {
  const float* preds     = (const float*)d_in[0];   // (NP, 25)
  const float* gt_boxes  = (const float*)d_in[1];   // (NT, 4)
  const int*   gt_labels = (const int*)d_in[2];     // (NT,)
  const float* priors    = (const float*)d_in[3];   // (NP, 4)
  float* out = (float*)d_out;

  char* w = (char*)d_ws;
  float*              best_iou = (float*)(w);                         // 4*NP
  int*                best_idx = (int*)(w + 4ull * NP);               // 4*NP
  unsigned*           mlkey    = (unsigned*)(w + 8ull * NP);          // 4*NP
  float*              ce_arr   = (float*)(w + 12ull * NP);            // 4*NP
  unsigned char*      posflag  = (unsigned char*)(w + 16ull * NP);    // NP
  unsigned long long* key64    = (unsigned long long*)(w + 17ull * NP); // 800B (8-aligned)
  char* s = w + 17ull * NP + 1024;
  int*      num_pos = (int*)(s + 0);
  unsigned* prefix  = (unsigned*)(s + 4);
  unsigned* krem    = (unsigned*)(s + 8);
  int*      tie_cnt = (int*)(s + 12);
  unsigned* hist    = (unsigned*)(s + 64);            // 256*4 = 1KB
  int*      tie_idx = (int*)(s + 64 + 1024);          // 4KB
  float* partial_base = (float*)(s + 64 + 1024 + 4096);           // 8KB
  float* partial_neg  = (float*)(s + 64 + 1024 + 4096 + 8192);    // 8KB

  k_init<<<1, 256, 0, stream>>>(key64, hist, num_pos, prefix, krem, tie_cnt);
  k_iou<<<NBLK, 256, 0, stream>>>(priors, gt_boxes, best_iou, best_idx, key64);
  k_override<<<1, 32, 0, stream>>>(key64, best_iou, best_idx);
  k_loss<<<NBLK, 256, 0, stream>>>(preds, gt_boxes, gt_labels, priors,
                                   best_iou, best_idx, mlkey, ce_arr, posflag,
                                   partial_base, num_pos);
  for (int pass = 0; pass < 4; ++pass) {
    k_hist<<<NBLK, 256, 0, stream>>>(mlkey, hist, prefix, pass);
    k_scan<<<1, 256, 0, stream>>>(hist, prefix, krem, num_pos, pass);
  }
  k_select<<<NBLK, 256, 0, stream>>>(mlkey, ce_arr, posflag, prefix,
                                     partial_neg, tie_cnt, tie_idx);
  k_final<<<1, 32, 0, stream>>>(partial_base, partial_neg, ce_arr, num_pos,
                                krem, tie_cnt, tie_idx, out);
}